// PI_DeepONet_15238543966928
// MI455X (gfx1250) — compile-verified
//
#include <hip/hip_runtime.h>
#include <hip/hip_bf16.h>

typedef float v2f __attribute__((ext_vector_type(2)));
typedef float v8f __attribute__((ext_vector_type(8)));

// ---------------------------------------------------------------------------
// Uniform cubic B-spline helpers. Grid: knots -2.2 + 0.4*i, i=0..11 (h=0.4).
// 8 basis functions B_0..B_7; on interval iv (local xi in [0,1)) only
// B_{iv-3..iv} are nonzero with the standard uniform-B-spline segments.
// ---------------------------------------------------------------------------
__device__ __forceinline__ void basis8(float u, float* B8) {
  float s  = (u + 2.2f) * 2.5f;           // (u - grid0) / h
  float fs = floorf(s);
  int   idx = (int)fs;
  bool  ok  = (idx >= 0) && (idx <= 10);
  float xi  = s - fs;
  float om  = 1.0f - xi;
  float xi2 = xi * xi, xi3 = xi2 * xi;
  float S0 = om * om * om * (1.0f / 6.0f);
  float S1 = (3.0f * xi3 - 6.0f * xi2 + 4.0f) * (1.0f / 6.0f);
  float S2 = (-3.0f * xi3 + 3.0f * xi2 + 3.0f * xi + 1.0f) * (1.0f / 6.0f);
  float S3 = xi3 * (1.0f / 6.0f);
#pragma unroll
  for (int k = 0; k < 8; ++k) {
    int   j = k - idx + 3;                 // B_{idx-3+j} = S_j
    float v = (j == 0) ? S0 : (j == 1) ? S1 : (j == 2) ? S2 : (j == 3) ? S3 : 0.0f;
    B8[k] = ok ? v : 0.0f;
  }
}

// ---------------------------------------------------------------------------
// Branch net: two [3,2,1] KANs over 1000 rows, evaluated 16 rows per wave
// with exact-f32 V_WMMA_F32_16X16X4_F32 GEMMs.
//   layer0: features = [u0,u1,u2, B(u0)x8, B(u1)x8, B(u2)x8]  -> 27 (pad 28)
//   layer1: features = [h0,h1, B(h0)x8, B(h1)x8]              -> 18 (pad 20)
// Weight fetches are branch-free (pointer/index cndmask, clamped in-bounds,
// zero-selected after) so all loads issue unconditionally up front and the
// WMMA accumulation chain runs with no interleaved memory stalls.
// ---------------------------------------------------------------------------
__device__ __forceinline__ float w_col_l0(int k, int n, const float* bw, const float* sw) {
  bool ok  = (n < 2) && (k < 27);
  int  nc  = (n < 2) ? n : 1;
  bool isb = (k < 3);
  int  ib  = nc * 3  + (isb ? k : 0);
  int  ks  = (k >= 3) ? (k - 3) : 0;  ks = (ks > 23) ? 23 : ks;
  int  is  = nc * 24 + ks;
  const float* p = isb ? bw : sw;
  int idx        = isb ? ib : is;
  float v = p[idx];                    // always in-bounds
  return ok ? v : 0.0f;
}
__device__ __forceinline__ float w_col_l1(int k, int n, const float* bw, const float* sw) {
  bool ok  = (n == 0) && (k < 18);
  bool isb = (k < 2);
  int  ks  = (k >= 2) ? (k - 2) : 0;  ks = (ks > 15) ? 15 : ks;
  const float* p = isb ? bw : sw;
  int idx        = isb ? (isb ? (k < 2 ? k : 1) : 0) : ks;
  float v = p[idx];
  return ok ? v : 0.0f;
}

__device__ v8f kan31_wmma(const float* U, int tile,
                          const float* bw0, const float* sw0,
                          const float* bw1, const float* sw1,
                          float* hbuf /* LDS, 32 floats */) {
  const int lane = threadIdx.x;
  const int m    = lane & 15;    // A: row M   / B,D: column N
  const int hi   = lane >> 4;    // half-wave select (K/M split)
  const int row  = tile * 16 + m;
  const bool rok = (row < 1000);
  const int rowc = rok ? row : 999;

  // ---- prefetch all weight fragments (unconditional loads, one wait) ----
  v2f B0f[7], B1f[5];
#pragma unroll
  for (int c = 0; c < 7; ++c) {
    int kb = 4 * c + 2 * hi;
    B0f[c].x = w_col_l0(kb + 0, m, bw0, sw0);
    B0f[c].y = w_col_l0(kb + 1, m, bw0, sw0);
  }
#pragma unroll
  for (int c = 0; c < 5; ++c) {
    int kb = 4 * c + 2 * hi;
    B1f[c].x = w_col_l1(kb + 0, m, bw1, sw1);
    B1f[c].y = w_col_l1(kb + 1, m, bw1, sw1);
  }

  // ---- features for this row (clamped load, zeroed if row invalid) ----
  float a = U[rowc * 3 + 0], b = U[rowc * 3 + 1], c0 = U[rowc * 3 + 2];
  a = rok ? a : 0.0f;  b = rok ? b : 0.0f;  c0 = rok ? c0 : 0.0f;
  float feat[28];
  feat[0] = a; feat[1] = b; feat[2] = c0;
  basis8(a,  feat + 3);
  basis8(b,  feat + 11);
  basis8(c0, feat + 19);
  feat[27] = 0.0f;
  if (!rok) {
#pragma unroll
    for (int k = 3; k < 27; ++k) feat[k] = 0.0f;
  }

  // Layer 0: D[m,n] = sum_k feat[m,k] * W0[k,n], K=28 as 7 chunks of 4.
  v8f acc = {0, 0, 0, 0, 0, 0, 0, 0};
#pragma unroll
  for (int c = 0; c < 7; ++c) {
    v2f A;
    A.x = hi ? feat[4 * c + 2] : feat[4 * c + 0];   // A: VGPR0 = K0/K2
    A.y = hi ? feat[4 * c + 3] : feat[4 * c + 1];   //    VGPR1 = K1/K3
    acc = __builtin_amdgcn_wmma_f32_16x16x4_f32(false, A, false, B0f[c], (short)0, acc, false, false);
  }

  // Scatter h0,h1 per row through LDS (D layout: lane n, vgpr v -> row v+8*hi).
  __syncthreads();
  if (m < 2) {
#pragma unroll
    for (int v = 0; v < 8; ++v) hbuf[(8 * hi + v) * 2 + m] = acc[v];
  }
  __syncthreads();

  float h0 = hbuf[m * 2 + 0], h1 = hbuf[m * 2 + 1];
  float f2[20];
  f2[0] = h0; f2[1] = h1;
  basis8(h0, f2 + 2);
  basis8(h1, f2 + 10);
  f2[18] = 0.0f; f2[19] = 0.0f;

  v8f acc2 = {0, 0, 0, 0, 0, 0, 0, 0};
#pragma unroll
  for (int c = 0; c < 5; ++c) {
    v2f A;
    A.x = hi ? f2[4 * c + 2] : f2[4 * c + 0];
    A.y = hi ? f2[4 * c + 3] : f2[4 * c + 1];
    acc2 = __builtin_amdgcn_wmma_f32_16x16x4_f32(false, A, false, B1f[c], (short)0, acc2, false, false);
  }
  return acc2;   // column 0 holds the kan output per row
}

__global__ void __launch_bounds__(32) branch_bc_kernel(
    const float* u1, const float* u2,
    const float* b10, const float* s10, const float* b11, const float* s11,
    const float* b20, const float* s20, const float* b21, const float* s21,
    float* bc) {
  __shared__ float hbuf1[32];
  __shared__ float hbuf2[32];
  const int tile = blockIdx.x;
  v8f d1 = kan31_wmma(u1, tile, b10, s10, b11, s11, hbuf1);
  v8f d2 = kan31_wmma(u2, tile, b20, s20, b21, s21, hbuf2);
  const int lane = threadIdx.x;
  const int m = lane & 15, hi = lane >> 4;
  if (m == 0) {                      // column N=0 lives in lanes 0 (rows 0-7) and 16 (rows 8-15)
#pragma unroll
    for (int v = 0; v < 8; ++v) {
      int row = tile * 16 + 8 * hi + v;
      if (row < 1000) bc[row] = d1[v] * d2[v];
    }
  }
}

// ---------------------------------------------------------------------------
// b0 = kan4(BC) (deterministic LDS tree reduction over the 1000-wide layer)
// + build the per-edge piecewise-cubic coefficient table for params5:
//   f(xi) on interval iv:  a0 + a1*xi + a2*xi^2 + a3*xi^3, with
//   a0=(c0+4c1+c2)/6, a1=(c2-c0)/2, a2=(c0-2c1+c2)/2, a3=(3(c1-c2)+c3-c0)/6
//   where c_j = sw[iv-3+j] (zero outside 0..7).
// ws layout: ws[0]=b0 ; float4 table at byte offset 64 (110 entries).
// ---------------------------------------------------------------------------
__global__ void __launch_bounds__(1024) b0_coef_kernel(
    const float* bc,
    const float* b40, const float* s40, const float* b41, const float* s41,
    const float* s50, const float* s51, const float* s52,
    float* ws) {
  __shared__ float p0[1024];
  __shared__ float p1[1024];
  const int tid = threadIdx.x;
  float a0 = 0.0f, a1 = 0.0f;
  if (tid < 1000) {
    float v = bc[tid];
    float B[8]; basis8(v, B);
    a0 = b40[tid] * v;
    a1 = b40[1000 + tid] * v;
#pragma unroll
    for (int k = 0; k < 8; ++k) {
      a0 = fmaf(s40[tid * 8 + k], B[k], a0);
      a1 = fmaf(s40[(1000 + tid) * 8 + k], B[k], a1);
    }
  }
  p0[tid] = a0; p1[tid] = a1;
  __syncthreads();
  for (int off = 512; off > 0; off >>= 1) {
    if (tid < off) { p0[tid] += p0[tid + off]; p1[tid] += p1[tid + off]; }
    __syncthreads();
  }
  if (tid == 0) {
    float h0 = p0[0], h1 = p1[0];
    float B0[8], B1[8]; basis8(h0, B0); basis8(h1, B1);
    float b0 = b41[0] * h0 + b41[1] * h1;
#pragma unroll
    for (int k = 0; k < 8; ++k) b0 = fmaf(s41[k], B0[k], fmaf(s41[8 + k], B1[k], b0));
    ws[0] = b0;
  }
  if (tid < 110) {                        // 10 edges x 11 intervals
    int e = tid / 11, iv = tid % 11;
    const float* sw = (e < 4) ? (s50 + e * 8) : (e < 8) ? (s51 + (e - 4) * 8) : (s52 + (e - 8) * 8);
    float c[4];
#pragma unroll
    for (int j = 0; j < 4; ++j) { int idx = iv - 3 + j; c[j] = (idx >= 0 && idx < 8) ? sw[idx] : 0.0f; }
    float4 A;
    A.x = (c[0] + 4.0f * c[1] + c[2]) * (1.0f / 6.0f);
    A.y = (c[2] - c[0]) * 0.5f;
    A.z = (c[0] - 2.0f * c[1] + c[2]) * 0.5f;
    A.w = (3.0f * (c[1] - c[2]) + c[3] - c[0]) * (1.0f / 6.0f);
    ((float4*)((char*)ws + 64))[tid] = A;
  }
}

// ---------------------------------------------------------------------------
// Trunk: 4.19M points, bandwidth-bound (48 MB @ 23.3 TB/s ~ 2.1 us floor).
// Forward-mode tuples (v, d/dx, d2/dx2, d/dt) through kan5 using the
// piecewise-cubic tables (Horner: 3/2/1 FMAs per edge quantity).
// ---------------------------------------------------------------------------
struct Tup { float v, x, xx, t; };

__device__ __forceinline__ void kan_layer(const float4* coef, int ebase,
                                          const float* bw, int nin, int nout,
                                          const Tup* in, Tup* out) {
#pragma unroll
  for (int j = 0; j < nout; ++j) out[j] = {0.0f, 0.0f, 0.0f, 0.0f};
#pragma unroll
  for (int i = 0; i < nin; ++i) {
    float u  = in[i].v;
    float s  = (u + 2.2f) * 2.5f;
    int   idx = (int)floorf(s);
    bool  ok  = (idx >= 0) && (idx <= 10);
    int   ic  = idx < 0 ? 0 : (idx > 10 ? 10 : idx);
    float xi  = s - (float)ic;
    float ux = in[i].x, uxx = in[i].xx, ut = in[i].t, ux2 = ux * ux;
#pragma unroll
    for (int j = 0; j < nout; ++j) {
      float4 a = coef[(ebase + j * nin + i) * 11 + ic];
      float p  = fmaf(fmaf(fmaf(a.w, xi, a.z), xi, a.y), xi, a.x);
      float d1 = fmaf(fmaf(3.0f * a.w, xi, 2.0f * a.z), xi, a.y) * 2.5f;     // * 1/h
      float d2 = fmaf(6.0f * a.w, xi, 2.0f * a.z) * 6.25f;                   // * 1/h^2
      p  = ok ? p  : 0.0f;
      d1 = ok ? d1 : 0.0f;
      d2 = ok ? d2 : 0.0f;
      float w  = bw[j * nin + i];
      float g1 = w + d1;
      out[j].v  += fmaf(w, u, p);
      out[j].x   = fmaf(g1, ux,  out[j].x);
      out[j].xx  = fmaf(g1, uxx, fmaf(d2, ux2, out[j].xx));
      out[j].t   = fmaf(g1, ut,  out[j].t);
    }
  }
}

__device__ __forceinline__ float point_eval(float x, float t, float b0,
                                            const float4* coef, const float* bw) {
  Tup u[2] = { {x, 1.0f, 0.0f, 0.0f}, {t, 0.0f, 0.0f, 1.0f} };
  Tup h1[2], h2[2], g[1];
  kan_layer(coef, 0, bw + 0, 2, 2, u,  h1);
  kan_layer(coef, 4, bw + 4, 2, 2, h1, h2);
  kan_layer(coef, 8, bw + 8, 2, 1, h2, g);
  const float C1 = 0.5f * 0.165856529f * 0.165856529f;
  const float C3 = 0.02563f;
  return b0 * (g[0].t - C1 * x * x * g[0].xx - C3 * x * g[0].x + C3 * g[0].v);
}

__global__ void __launch_bounds__(256) residual_kernel(
    const float* __restrict__ x, const float* __restrict__ t,
    const float* __restrict__ ws,
    const float* __restrict__ bw50, const float* __restrict__ bw51,
    const float* __restrict__ bw52,
    float* __restrict__ out, int N) {
  __shared__ float4 s_coef[110];
  __shared__ float  s_bw[10];
  __shared__ float  s_b0;
  const int tid = threadIdx.x;
  const float4* gcoef = (const float4*)((const char*)ws + 64);
  if (tid < 110) s_coef[tid] = gcoef[tid];
  if (tid >= 128 && tid < 138) {
    int q = tid - 128;
    s_bw[q] = (q < 4) ? bw50[q] : (q < 8) ? bw51[q - 4] : bw52[q - 8];
  }
  if (tid == 138) s_b0 = ws[0];
  __syncthreads();
  float b0 = s_b0;
  int g = blockIdx.x * blockDim.x + tid;
  int base = g * 4;
  if (base + 3 < N) {
    float4 xv = ((const float4*)x)[g];
    float4 tv = ((const float4*)t)[g];
    float4 r;
    r.x = point_eval(xv.x, tv.x, b0, s_coef, s_bw);
    r.y = point_eval(xv.y, tv.y, b0, s_coef, s_bw);
    r.z = point_eval(xv.z, tv.z, b0, s_coef, s_bw);
    r.w = point_eval(xv.w, tv.w, b0, s_coef, s_bw);
    ((float4*)out)[g] = r;
  } else {
    for (int i = base; i < N; ++i) out[i] = point_eval(x[i], t[i], b0, s_coef, s_bw);
  }
}

// ---------------------------------------------------------------------------
// d_in order (leaves of setup_inputs): u1,u2,u_s1,u_s2,x,t,
//   p1(bw0,sw0,bw1,sw1), p2(...), p4(...), p5(bw0,sw0,bw1,sw1,bw2,sw2)
// ---------------------------------------------------------------------------
extern "C" void kernel_launch(void* const* d_in, const int* in_sizes, int n_in,
                              void* d_out, int out_size, void* d_ws, size_t ws_size,
                              hipStream_t stream) {
  const float* u1  = (const float*)d_in[0];
  const float* u2  = (const float*)d_in[1];
  const float* x   = (const float*)d_in[4];
  const float* t   = (const float*)d_in[5];
  const float* b10 = (const float*)d_in[6];
  const float* s10 = (const float*)d_in[7];
  const float* b11 = (const float*)d_in[8];
  const float* s11 = (const float*)d_in[9];
  const float* b20 = (const float*)d_in[10];
  const float* s20 = (const float*)d_in[11];
  const float* b21 = (const float*)d_in[12];
  const float* s21 = (const float*)d_in[13];
  const float* b40 = (const float*)d_in[14];
  const float* s40 = (const float*)d_in[15];
  const float* b41 = (const float*)d_in[16];
  const float* s41 = (const float*)d_in[17];
  const float* b50 = (const float*)d_in[18];
  const float* s50 = (const float*)d_in[19];
  const float* b51 = (const float*)d_in[20];
  const float* s51 = (const float*)d_in[21];
  const float* b52 = (const float*)d_in[22];
  const float* s52 = (const float*)d_in[23];

  float* ws = (float*)d_ws;
  float* bc = ws + 1024;                 // BC[1000] scratch
  int N = in_sizes[4];

  branch_bc_kernel<<<63, 32, 0, stream>>>(u1, u2, b10, s10, b11, s11,
                                          b20, s20, b21, s21, bc);
  b0_coef_kernel<<<1, 1024, 0, stream>>>(bc, b40, s40, b41, s41, s50, s51, s52, ws);

  int threads = 256;
  int blocks = (N + threads * 4 - 1) / (threads * 4);
  residual_kernel<<<blocks, threads, 0, stream>>>(x, t, ws, b50, b51, b52,
                                                  (float*)d_out, N);
}